// SimpleGCN_54949811585248
// MI455X (gfx1250) — compile-verified
//
#include <hip/hip_runtime.h>

#define DIMS 128

typedef __attribute__((ext_vector_type(2))) float v2f;
typedef __attribute__((ext_vector_type(8))) float v8f;

// ---------------------------------------------------------------- utilities
__global__ void k_zero(float* __restrict__ p, int n) {
    int i = blockIdx.x * blockDim.x + threadIdx.x;
    if (i < n) p[i] = 0.0f;
}

// deg[i] = (#edges with src==i) + (#edges with dst==i)
__global__ void k_count_deg(const int* __restrict__ src, const int* __restrict__ dst,
                            float* __restrict__ deg, int E) {
    int e = blockIdx.x * blockDim.x + threadIdx.x;
    if (e < E) {
        atomicAdd(&deg[src[e]], 1.0f);
        atomicAdd(&deg[dst[e]], 1.0f);
    }
}

// dis[i] = clip(deg_total, 1)^-0.5, deg_total = in+out+1 (self loop net +1)
__global__ void k_dis(const float* __restrict__ deg, float* __restrict__ dis, int n) {
    int i = blockIdx.x * blockDim.x + threadIdx.x;
    if (i < n) {
        float d = fmaxf(deg[i] + 1.0f, 1.0f);
        dis[i] = 1.0f / sqrtf(d);
    }
}

// Self-loop term initializes agg (so the edge kernel only needs atomics):
// agg[i][:] = dis[i]^2 * H[i][:]
__global__ void k_init_agg(const float* __restrict__ H, const float* __restrict__ dis,
                           float* __restrict__ agg, int N) {
    int t = blockIdx.x * blockDim.x + threadIdx.x;
    int total = N * (DIMS / 4);
    if (t < total) {
        int node = t >> 5;          // DIMS/4 == 32 chunks per node
        int q    = t & 31;
        float s = dis[node];
        s = s * s;
        float4 h = ((const float4*)(H + (size_t)node * DIMS))[q];
        float4 o;
        o.x = s * h.x; o.y = s * h.y; o.z = s * h.z; o.w = s * h.w;
        ((float4*)(agg + (size_t)node * DIMS))[q] = o;
    }
}

// Edge scatter: agg[src] += dis[src]*dis[dst] * H[dst].
// 32 lanes per edge, float4 gather of H[dst], 4x global_atomic_add_f32 each.
__global__ void k_spmm(const int* __restrict__ src, const int* __restrict__ dst,
                       const float* __restrict__ dis, const float* __restrict__ H,
                       float* __restrict__ agg, int E) {
    long long t = (long long)blockIdx.x * blockDim.x + threadIdx.x;
    int e    = (int)(t >> 5);
    int lane = (int)(t & 31);
    if (e < E) {
        int s = src[e];
        int d = dst[e];
        float v = dis[s] * dis[d];
        float4 h = ((const float4*)(H + (size_t)d * DIMS))[lane];
        float* out = agg + (size_t)s * DIMS + lane * 4;
        atomicAdd(out + 0, v * h.x);
        atomicAdd(out + 1, v * h.y);
        atomicAdd(out + 2, v * h.z);
        atomicAdd(out + 3, v * h.w);
    }
}

// OUT = relu(A @ W^T), A: N x 128, W: 128 x 128 (row-major), OUT: N x 128.
// Block = 256 threads = 8 waves. Block b owns node rows [16b, 16b+16);
// wave w owns output cols [16w, 16w+16). A-tile staged in padded LDS.
// WMMA: V_WMMA_F32_16X16X4_F32, K stepped by 4, 32 issues per wave.
__global__ void __launch_bounds__(256) k_gemm_relu(const float* __restrict__ A,
                                                   const float* __restrict__ W,
                                                   float* __restrict__ OUT, int N) {
    __shared__ float lA[16][DIMS + 4];   // pad 4 floats -> conflict-free K reads

    const int tid  = threadIdx.x;
    const int wave = tid >> 5;           // 0..7 -> column tile
    const int lane = tid & 31;
    const int row0 = blockIdx.x * 16;

    // Cooperative, coalesced load of the 16x128 A tile (512 float4, 2/thread).
    const float4* a4 = (const float4*)(A + (size_t)row0 * DIMS);
    for (int f = tid; f < 16 * (DIMS / 4); f += 256) {
        int r  = f >> 5;                 // row within tile
        int c4 = f & 31;                 // float4 column
        float4 v;
        if (row0 + r < N) {
            v = a4[r * (DIMS / 4) + c4];
        } else {
            v.x = v.y = v.z = v.w = 0.0f;
        }
        lA[r][c4 * 4 + 0] = v.x;
        lA[r][c4 * 4 + 1] = v.y;
        lA[r][c4 * 4 + 2] = v.z;
        lA[r][c4 * 4 + 3] = v.w;
    }
    __syncthreads();

    const int col0 = wave * 16;
    const int half = lane >> 4;          // 0: K{0,1}; 1: K{2,3}
    const int l    = lane & 15;          // M index for A, N index for B

    // B[kk][n] = W^T[kk][n] = W[col0+n][k+kk]; this lane streams row (col0+l) of W.
    const float* wrow = W + (size_t)(col0 + l) * DIMS;

    v8f c = {};
    for (int k = 0; k < DIMS; k += 4) {
        // A fragment per ISA layout: VGPR0 = K0 (lanes 0-15) / K2 (lanes 16-31),
        //                            VGPR1 = K1 / K3.
        v2f a = *(const v2f*)&lA[l][k + 2 * half];
        // B fragment, mirrored layout (K along VGPRs, N along lanes).
        v2f b = *(const v2f*)(wrow + k + 2 * half);
        c = __builtin_amdgcn_wmma_f32_16x16x4_f32(
                /*neg_a=*/false, a, /*neg_b=*/false, b,
                /*c_mod=*/(short)0, c, /*reuse_a=*/false, /*reuse_b=*/false);
    }

    // C/D layout: VGPR r = M=r (lanes 0-15) / M=8+r (lanes 16-31), N = lane%16.
    #pragma unroll
    for (int r = 0; r < 8; ++r) {
        int row = row0 + r + 8 * half;
        if (row < N) {
            OUT[(size_t)row * DIMS + col0 + l] = fmaxf(c[r], 0.0f);
        }
    }
}

// ---------------------------------------------------------------- launcher
extern "C" void kernel_launch(void* const* d_in, const int* in_sizes, int n_in,
                              void* d_out, int out_size, void* d_ws, size_t ws_size,
                              hipStream_t stream) {
    const float* emb = (const float*)d_in[0];
    const float* W1  = (const float*)d_in[1];
    const float* W2  = (const float*)d_in[2];
    const int*   edg = (const int*)d_in[3];

    const int N = in_sizes[0] / DIMS;
    const int E = in_sizes[3] / 2;
    const int* src = edg;          // edges[0]
    const int* dst = edg + E;      // edges[1]

    float* ws  = (float*)d_ws;
    float* deg = ws;                                 // N
    float* dis = ws + N;                             // N
    float* agg = ws + 2 * (size_t)N;                 // N*DIMS
    float* H1  = agg + (size_t)N * DIMS;             // N*DIMS

    const int T = 256;

    // --- normalized adjacency values ---
    k_zero<<<(N + T - 1) / T, T, 0, stream>>>(deg, N);
    k_count_deg<<<(E + T - 1) / T, T, 0, stream>>>(src, dst, deg, E);
    k_dis<<<(N + T - 1) / T, T, 0, stream>>>(deg, dis, N);

    const int initBlocks = (N * (DIMS / 4) + T - 1) / T;
    const unsigned spmmBlocks = (unsigned)(((long long)E * 32 + T - 1) / T);
    const int gemmBlocks = (N + 15) / 16;

    // --- layer 1: emb -> H1 ---
    k_init_agg<<<initBlocks, T, 0, stream>>>(emb, dis, agg, N);
    k_spmm<<<spmmBlocks, T, 0, stream>>>(src, dst, dis, emb, agg, E);
    k_gemm_relu<<<gemmBlocks, T, 0, stream>>>(agg, W1, H1, N);

    // --- layer 2: H1 -> d_out ---
    k_init_agg<<<initBlocks, T, 0, stream>>>(H1, dis, agg, N);
    k_spmm<<<spmmBlocks, T, 0, stream>>>(src, dst, dis, H1, agg, E);
    k_gemm_relu<<<gemmBlocks, T, 0, stream>>>(agg, W2, (float*)d_out, N);
}